// DynamicPooling_26946624815393
// MI455X (gfx1250) — compile-verified
//
#include <hip/hip_runtime.h>
#include <math.h>

// Problem constants (from reference): B=64, N=4096, D=256, 3 routing iters.
#define BB 64
#define NN 4096
#define DD 256
#define N_BPB 16                   // blocks per batch for big passes
#define ROWS_PB (NN / N_BPB)       // 256 rows per block
#define TROWS 16                   // tile rows (WMMA M)
#define NTILES (ROWS_PB / TROWS)   // 16 tiles per block
#define YSTR 260                   // LDS row stride (floats): 16B-aligned rows, bank-skewed

typedef __attribute__((ext_vector_type(2))) float v2f;
typedef __attribute__((ext_vector_type(8))) float v8f;

// Async-to-LDS builtin pointee type (from hipcc diagnostic):
//   param0: int __vector(4) __device__ * (AS1), param1: AS3 equivalent
typedef int v4i __attribute__((vector_size(16)));
typedef v4i __attribute__((address_space(1))) as1_v4i;
typedef v4i __attribute__((address_space(3))) as3_v4i;

#if __has_builtin(__builtin_amdgcn_global_load_async_to_lds_b128)
#define HAVE_ASYNC 1
#else
#define HAVE_ASYNC 0
#endif

__device__ __forceinline__ void wait_async_imm4() {
#if HAVE_ASYNC
#if __has_builtin(__builtin_amdgcn_s_wait_asynccnt)
  __builtin_amdgcn_s_wait_asynccnt(4);
#else
  asm volatile("s_wait_asynccnt 4" ::: "memory");
#endif
#endif
}
__device__ __forceinline__ void wait_async_imm0() {
#if HAVE_ASYNC
#if __has_builtin(__builtin_amdgcn_s_wait_asynccnt)
  __builtin_amdgcn_s_wait_asynccnt(0);
#else
  asm volatile("s_wait_asynccnt 0" ::: "memory");
#endif
#endif
}

__device__ __forceinline__ void atomicMaxF(float* addr, float val) {
  unsigned int* ai = (unsigned int*)addr;
  unsigned int old = __float_as_uint(*addr);
  while (__uint_as_float(old) < val) {
    unsigned int assumed = old;
    old = atomicCAS(ai, assumed, __float_as_uint(val));
    if (old == assumed) break;
  }
}

// Stage one 16x256 f32 tile of x into LDS (row stride YSTR floats).
// 256 threads, 4 x b128 per thread; coalesced (consecutive tid -> consecutive 16B).
__device__ __forceinline__ void stage_tile(const float* __restrict__ gsrc,
                                           float* lbuf, int tid) {
#pragma unroll
  for (int i = 0; i < 4; ++i) {
    const int s = i * 256 + tid;
    const int r = s >> 6;        // 64 float4 per row
    const int c4 = s & 63;
    const float* gp = gsrc + (size_t)r * DD + (size_t)c4 * 4;
    float* lp = lbuf + r * YSTR + c4 * 4;
#if HAVE_ASYNC
    __builtin_amdgcn_global_load_async_to_lds_b128((as1_v4i*)gp, (as3_v4i*)lp, 0, 0);
#else
    *(float4*)lp = *(const float4*)gp;
#endif
  }
}

// ---------- init: zero state ----------
__global__ void init_kernel(float* __restrict__ sigma_acc, float* __restrict__ z_acc,
                            float* __restrict__ mx, float* __restrict__ shiftv,
                            float* __restrict__ b) {
  const int i = blockIdx.x * blockDim.x + threadIdx.x;
  if (i < BB * NN) b[i] = 0.f;
  if (i < BB * DD) sigma_acc[i] = 0.f;
  if (i < BB) { z_acc[i] = 0.f; mx[i] = 0.0f; shiftv[i] = 0.f; }
}

// ---------- pass 0: sigma_1 partial = sum_n m_n * x_n  (uniform weights) ----------
__global__ void __launch_bounds__(256) pass0_kernel(const float* __restrict__ x,
                                                    const float* __restrict__ m,
                                                    float* __restrict__ sigma_acc,
                                                    float* __restrict__ z_acc) {
  const int tid = threadIdx.x;
  const int batch = blockIdx.x / N_BPB;
  const int nbase = (blockIdx.x % N_BPB) * ROWS_PB;
  const float* xblk = x + ((size_t)batch * NN + nbase) * DD;
  const float* mblk = m + (size_t)batch * NN + nbase;
  float acc = 0.f;
  for (int n = 0; n < ROWS_PB; ++n) {
    if (n + 4 < ROWS_PB) __builtin_prefetch(xblk + (size_t)(n + 4) * DD + tid, 0, 0);
    acc += mblk[n] * xblk[(size_t)n * DD + tid];
  }
  atomicAdd(&sigma_acc[batch * DD + tid], acc);
  if (tid == 0) atomicAdd(&z_acc[batch], (float)ROWS_PB);
}

// ---------- fused pass: b_n += m_n*<x_n,s>; then (unless last) accumulate
//            un-normalized softmax stats for the NEXT sigma with updated b ----------
__global__ void __launch_bounds__(256) froute_kernel(
    const float* __restrict__ x, const float* __restrict__ m,
    const float* __restrict__ s_in, float* __restrict__ b,
    float* __restrict__ sigma_acc, float* __restrict__ z_acc,
    float* __restrict__ mx, const float* __restrict__ shiftv, int last) {
  __shared__ __align__(16) float yt[2][TROWS * YSTR];
  __shared__ float sl[DD];
  __shared__ float el[TROWS];

  const int tid = threadIdx.x;
  const int lane = tid & 31;
  const int wave = tid >> 5;
  const int batch = blockIdx.x / N_BPB;
  const int nbase = (blockIdx.x % N_BPB) * ROWS_PB;

  sl[tid] = s_in[batch * DD + tid];
  const float C = shiftv[batch];

  float acc = 0.f, zpart = 0.f, lmax = -3.0e38f;
  const float* xblk = x + ((size_t)batch * NN + nbase) * DD;

  stage_tile(xblk, yt[0], tid);

  for (int t = 0; t < NTILES; ++t) {
    const int cur = t & 1;
    if (t + 1 < NTILES) {
      stage_tile(xblk + (size_t)(t + 1) * TROWS * DD, yt[(t + 1) & 1], tid);
      wait_async_imm4();   // older tile's 4 async ops complete (in-order), next 4 in flight
    } else {
      wait_async_imm0();
    }
    __syncthreads();

    if (wave == 0) {
      // C[mrow, j] = sum_k x[mrow,k] * s[k]  via 64x V_WMMA_F32_16X16X4_F32
      v8f c = {0.f, 0.f, 0.f, 0.f, 0.f, 0.f, 0.f, 0.f};
      const int ksel = (lane < 16) ? 0 : 2;
      const int lm = lane & 15;
      const float* yrow = &yt[cur][lm * YSTR];
      for (int k0 = 0; k0 < DD; k0 += 4) {
        v2f a; a.x = yrow[k0 + ksel]; a.y = yrow[k0 + ksel + 1];
        v2f bf; bf.x = sl[k0 + ksel]; bf.y = sl[k0 + ksel + 1];
        c = __builtin_amdgcn_wmma_f32_16x16x4_f32(false, a, false, bf,
                                                  (short)0, c, false, false);
      }
      // C layout: lane 0 holds rows 0..7 (vgpr r), lane 16 holds rows 8..15
      if (lane == 0 || lane == 16) {
        const int rb = (lane == 0) ? 0 : 8;
#pragma unroll
        for (int r = 0; r < 8; ++r) {
          const size_t row = (size_t)batch * NN + nbase + t * TROWS + rb + r;
          const float dv = c[r];
          const float mv = m[row];
          const float bnew = b[row] + mv * dv;   // y = m*x -> <y,s> = m*<x,s>
          b[row] = bnew;
          const float logit = last ? bnew : (mv * bnew);
          lmax = fmaxf(lmax, logit);
          if (!last) {
            const float e = __expf(logit - C);   // shift-invariant; C = lagged max
            zpart += e;
            el[rb + r] = e * mv;                 // fold m into weight for u-accum
          }
        }
      }
    }
    __syncthreads();
    if (!last) {
      const float* ycol = &yt[cur][tid];
#pragma unroll
      for (int r = 0; r < TROWS; ++r) acc += el[r] * ycol[r * YSTR];
    }
    __syncthreads();  // protect buffer reuse by next stage
  }

  if (!last) atomicAdd(&sigma_acc[batch * DD + tid], acc);
  if (wave == 0 && (lane == 0 || lane == 16)) {
    atomicMaxF(&mx[batch], lmax);
    if (!last) atomicAdd(&z_acc[batch], zpart);
  }
}

// ---------- squash: s = squash(u/Z); rotate shift <- mx; reset accumulators ----------
__global__ void __launch_bounds__(256) squash_kernel(
    float* __restrict__ sigma_acc, float* __restrict__ z_acc,
    float* __restrict__ s_out, float* __restrict__ mx, float* __restrict__ shiftv,
    float* __restrict__ out_s) {
  __shared__ float red[256];
  const int batch = blockIdx.x, tid = threadIdx.x;
  const float Z = z_acc[batch];
  const float sig = sigma_acc[batch * DD + tid] / Z;
  red[tid] = sig * sig;
  __syncthreads();
  for (int s2 = 128; s2 > 0; s2 >>= 1) {
    if (tid < s2) red[tid] += red[tid + s2];
    __syncthreads();
  }
  const float n2 = red[0];
  const float nrm = sqrtf(n2);
  const float sv = (n2 / (1.f + n2) / (nrm + 1e-8f)) * sig;
  s_out[batch * DD + tid] = sv;
  if (out_s) out_s[batch * DD + tid] = sv;
  sigma_acc[batch * DD + tid] = 0.f;
  if (tid == 0) { shiftv[batch] = mx[batch]; mx[batch] = -3.0e38f; z_acc[batch] = 0.f; }
}

// ---------- final softmax over b: Z = sum exp(b - M), then w ----------
__global__ void __launch_bounds__(256) zfin_kernel(const float* __restrict__ b,
                                                   const float* __restrict__ mx,
                                                   float* __restrict__ z_acc) {
  __shared__ float red[256];
  const int batch = blockIdx.x / N_BPB;
  const int i0 = (blockIdx.x % N_BPB) * ROWS_PB;
  const int tid = threadIdx.x;
  red[tid] = __expf(b[(size_t)batch * NN + i0 + tid] - mx[batch]);
  __syncthreads();
  for (int s2 = 128; s2 > 0; s2 >>= 1) {
    if (tid < s2) red[tid] += red[tid + s2];
    __syncthreads();
  }
  if (tid == 0) atomicAdd(&z_acc[batch], red[0]);
}

__global__ void __launch_bounds__(256) wout_kernel(const float* __restrict__ b,
                                                   const float* __restrict__ mx,
                                                   const float* __restrict__ z_acc,
                                                   float* __restrict__ w) {
  const int i = blockIdx.x * 256 + threadIdx.x;
  const int batch = i >> 12;  // N = 4096
  w[i] = __expf(b[i] - mx[batch]) / z_acc[batch];
}

extern "C" void kernel_launch(void* const* d_in, const int* in_sizes, int n_in,
                              void* d_out, int out_size, void* d_ws, size_t ws_size,
                              hipStream_t stream) {
  (void)in_sizes; (void)n_in; (void)out_size; (void)ws_size;
  const float* x = (const float*)d_in[0];
  const float* m = (const float*)d_in[1];
  float* out = (float*)d_out;

  float* ws = (float*)d_ws;
  float* sigma_acc = ws;                  // B*D
  float* s_buf     = ws + BB * DD;        // B*D
  float* z_acc     = ws + 2 * BB * DD;    // B
  float* mx        = z_acc + BB;          // B
  float* shiftv    = mx + BB;             // B
  float* b         = shiftv + BB;         // B*N
  float* w_out = out;                     // [B*N]
  float* s_out = out + BB * NN;           // [B*D]

  init_kernel<<<BB * NN / 256, 256, 0, stream>>>(sigma_acc, z_acc, mx, shiftv, b);
  // iter 1: uniform weights
  pass0_kernel<<<BB * N_BPB, 256, 0, stream>>>(x, m, sigma_acc, z_acc);
  squash_kernel<<<BB, 256, 0, stream>>>(sigma_acc, z_acc, s_buf, mx, shiftv, nullptr);
  // iter 1->2 fused: b update + sigma_2 stats
  froute_kernel<<<BB * N_BPB, 256, 0, stream>>>(x, m, s_buf, b, sigma_acc, z_acc, mx, shiftv, 0);
  squash_kernel<<<BB, 256, 0, stream>>>(sigma_acc, z_acc, s_buf, mx, shiftv, nullptr);
  // iter 2->3 fused: b update + sigma_3 stats
  froute_kernel<<<BB * N_BPB, 256, 0, stream>>>(x, m, s_buf, b, sigma_acc, z_acc, mx, shiftv, 0);
  squash_kernel<<<BB, 256, 0, stream>>>(sigma_acc, z_acc, s_buf, mx, shiftv, s_out);  // s output
  // final b update + max(b) for output softmax
  froute_kernel<<<BB * N_BPB, 256, 0, stream>>>(x, m, s_buf, b, sigma_acc, z_acc, mx, shiftv, 1);
  zfin_kernel<<<BB * N_BPB, 256, 0, stream>>>(b, mx, z_acc);
  wout_kernel<<<BB * NN / 256, 256, 0, stream>>>(b, mx, z_acc, w_out);
}